// FMGNN_171798692208
// MI455X (gfx1250) — compile-verified
//
#include <hip/hip_runtime.h>
#include <hip/hip_bf16.h>

// ---------------------------------------------------------------------------
// CDNA5 (gfx1250) implementation of the FM-GNN compound/protein model.
// Dominant work: 3x (4000x4000)@(4000x128) f32 GEMMs -> V_WMMA_F32_16X16X4_F32
// with 4-way split-K for occupancy (1000 waves) + deterministic reduction.
// ---------------------------------------------------------------------------

typedef __attribute__((ext_vector_type(2))) float v2f;
typedef __attribute__((ext_vector_type(8))) float v8f;

#define NA_ 4000
#define NS_ 4000
#define D_  128
#define NF_ 50000
#define NW_ 30000
#define SPLITK_ 4

// ---------------------------------------------------------------------------
// Embedding gather: out[n, 0:128] = table[idx[n], 0:128]
// ---------------------------------------------------------------------------
__global__ __launch_bounds__(256) void k_gather_embed(
    const float* __restrict__ table, const int* __restrict__ idx,
    float* __restrict__ out, int N) {
  int t = blockIdx.x * 256 + threadIdx.x;
  if (t >= N * D_) return;
  int n = t >> 7, d = t & 127;
  out[t] = table[(size_t)idx[n] * D_ + d];
}

// ---------------------------------------------------------------------------
// out[m,n] = relu( sum_k X[m,k] * W[n,k] + b[n] ),  X:[M,128], W:[128,128]
// One wave -> 16x64 strip (4 WMMA accumulators); blockIdx.y picks N half.
// ---------------------------------------------------------------------------
__global__ __launch_bounds__(128) void k_gemm_xw_relu(
    const float* __restrict__ X, const float* __restrict__ W,
    const float* __restrict__ bias, float* __restrict__ out, int M) {
  int wave = blockIdx.x * (blockDim.x >> 5) + (threadIdx.x >> 5);
  int lane = threadIdx.x & 31;
  int m0 = wave * 16;
  if (m0 >= M) return;                 // whole-wave exit: EXEC stays all-ones
  int ml = lane & 15;                  // A: row M;  B/C: col N (within tile)
  int kh = (lane >> 4) << 1;           // K sub-pair selected by half-wave
  int ntb = blockIdx.y * 4;            // N-tile base (0 or 4)

  v8f acc[4] = {};
  const float* xrow = X + (size_t)(m0 + ml) * D_;

  for (int k0 = 0; k0 < D_; k0 += 4) {
    int kb = k0 + kh;
    v2f a;
    a.x = xrow[kb];
    a.y = xrow[kb + 1];
#pragma unroll
    for (int nt = 0; nt < 4; ++nt) {
      int n = (ntb + nt) * 16 + ml;
      v2f b;
      b.x = W[n * D_ + kb];            // B[k][n] = W[n][k]
      b.y = W[n * D_ + kb + 1];
      acc[nt] = __builtin_amdgcn_wmma_f32_16x16x4_f32(
          false, a, false, b, (short)0, acc[nt], false, false);
    }
  }
  // C/D layout: VGPR r -> row (r + 8*(lane>>4)), lane&15 -> col
  int rb = (lane >> 4) * 8;
#pragma unroll
  for (int nt = 0; nt < 4; ++nt) {
    int n = (ntb + nt) * 16 + ml;
    float bn = bias[n];
#pragma unroll
    for (int r = 0; r < 8; ++r) {
      int m = m0 + rb + r;
      out[(size_t)m * D_ + n] = fmaxf(acc[nt][r] + bn, 0.0f);
    }
  }
}

// ---------------------------------------------------------------------------
// Split-K partial: P[split][m,n] = sum_{k in chunk} A[m,k]*B[k,n]
// A:[M,K] adjacency, B:[K,128].  blockIdx.y = split index.
// ---------------------------------------------------------------------------
__global__ __launch_bounds__(128) void k_gemm_adj_partial(
    const float* __restrict__ A, const float* __restrict__ B,
    float* __restrict__ P, int M, int K, int Kc) {
  int wave = blockIdx.x * (blockDim.x >> 5) + (threadIdx.x >> 5);
  int lane = threadIdx.x & 31;
  int m0 = wave * 16;
  if (m0 >= M) return;
  int ml = lane & 15;
  int kh = (lane >> 4) << 1;
  int kbeg = blockIdx.y * Kc;
  int kend = kbeg + Kc;

  v8f acc[8] = {};
  const float* arow = A + (size_t)(m0 + ml) * K;

  for (int k0 = kbeg; k0 < kend; k0 += 4) {
    int kb = k0 + kh;
    v2f a;
    a.x = arow[kb];
    a.y = arow[kb + 1];
    const float* b0 = B + (size_t)kb * D_;
#pragma unroll
    for (int nt = 0; nt < 8; ++nt) {
      int n = nt * 16 + ml;
      v2f b;
      b.x = b0[n];                     // B[kb  ][n]  (16-lane coalesced)
      b.y = b0[D_ + n];                // B[kb+1][n]
      acc[nt] = __builtin_amdgcn_wmma_f32_16x16x4_f32(
          false, a, false, b, (short)0, acc[nt], false, false);
    }
  }
  float* out = P + (size_t)blockIdx.y * (size_t)M * D_;
  int rb = (lane >> 4) * 8;
#pragma unroll
  for (int nt = 0; nt < 8; ++nt) {
    int n = nt * 16 + ml;
#pragma unroll
    for (int r = 0; r < 8; ++r) {
      out[(size_t)(m0 + rb + r) * D_ + n] = acc[nt][r];
    }
  }
}

// C[i] += P[0][i] + P[1][i] + ... (fixed order -> deterministic)
__global__ __launch_bounds__(256) void k_reduce_splitk(
    float* __restrict__ C, const float* __restrict__ P, int total) {
  int i = blockIdx.x * 256 + threadIdx.x;
  if (i >= total) return;
  float acc = C[i];
#pragma unroll
  for (int s = 0; s < SPLITK_; ++s) acc += P[(size_t)s * total + i];
  C[i] = acc;
}

// ---------------------------------------------------------------------------
// Fallback (workspace too small for split-K): C[m,n] += sum_k A[m,k]*B[k,n]
// ---------------------------------------------------------------------------
__global__ __launch_bounds__(128) void k_gemm_adj(
    const float* __restrict__ A, const float* __restrict__ B,
    float* __restrict__ C, int M, int K) {
  int wave = blockIdx.x * (blockDim.x >> 5) + (threadIdx.x >> 5);
  int lane = threadIdx.x & 31;
  int m0 = wave * 16;
  if (m0 >= M) return;
  int ml = lane & 15;
  int kh = (lane >> 4) << 1;

  v8f acc[8] = {};
  const float* arow = A + (size_t)(m0 + ml) * K;

  for (int k0 = 0; k0 < K; k0 += 4) {
    int kb = k0 + kh;
    v2f a;
    a.x = arow[kb];
    a.y = arow[kb + 1];
    const float* b0 = B + (size_t)kb * D_;
#pragma unroll
    for (int nt = 0; nt < 8; ++nt) {
      int n = nt * 16 + ml;
      v2f b;
      b.x = b0[n];
      b.y = b0[D_ + n];
      acc[nt] = __builtin_amdgcn_wmma_f32_16x16x4_f32(
          false, a, false, b, (short)0, acc[nt], false, false);
    }
  }
  int rb = (lane >> 4) * 8;
#pragma unroll
  for (int nt = 0; nt < 8; ++nt) {
    int n = nt * 16 + ml;
#pragma unroll
    for (int r = 0; r < 8; ++r) {
      size_t o = (size_t)(m0 + rb + r) * D_ + n;
      C[o] = C[o] + acc[nt][r];
    }
  }
}

// ---------------------------------------------------------------------------
// 11x11 same-padded conv over [S,128] plane + bias + relu (cross-correlation)
// ---------------------------------------------------------------------------
__global__ __launch_bounds__(256) void k_conv11_relu(
    const float* __restrict__ in, float* __restrict__ out,
    const float* __restrict__ wall, const float* __restrict__ ball,
    int layer, int S) {
  __shared__ float tile[18][43];       // (8+10) x (32+10), +1 pad
  const float* w = wall + layer * 121;
  float bias = ball[layer];
  int d0 = blockIdx.x * 32;
  int s0 = blockIdx.y * 8;
  for (int i = threadIdx.x; i < 18 * 42; i += 256) {
    int ls = i / 42, ld = i % 42;
    int s = s0 + ls - 5, d = d0 + ld - 5;
    float v = 0.0f;
    if (s >= 0 && s < S && d >= 0 && d < D_) v = in[(size_t)s * D_ + d];
    tile[ls][ld] = v;
  }
  __syncthreads();
  int tx = threadIdx.x & 31;           // d within tile
  int ty = threadIdx.x >> 5;           // s within tile
  float acc = bias;
#pragma unroll
  for (int i = 0; i < 11; ++i)
#pragma unroll
    for (int j = 0; j < 11; ++j)
      acc += w[i * 11 + j] * tile[ty + i][tx + j];
  int s = s0 + ty, d = d0 + tx;
  if (s < S) out[(size_t)s * D_ + d] = fmaxf(acc, 0.0f);
}

// ---------------------------------------------------------------------------
// cat init: [0:128)=0, [128:256)=bc, [256:384)=0, [384:512)=bp
// (FM interaction term p1*p1 - p2 is exactly zero for one-hot inputs.)
// ---------------------------------------------------------------------------
__global__ __launch_bounds__(512) void k_init_cat(
    float* __restrict__ cat, const float* __restrict__ bc,
    const float* __restrict__ bp) {
  int t = threadIdx.x;
  float v = 0.0f;
  if (t >= 128 && t < 256) v = bc[t - 128];
  else if (t >= 384) v = bp[t - 384];
  cat[t] = v;
}

// out128[d] += scale * sum_s X[s,d]
__global__ __launch_bounds__(128) void k_colmean(
    const float* __restrict__ X, int S, float scale,
    float* __restrict__ out128) {
  int d = threadIdx.x;
  int chunk = (S + gridDim.x - 1) / gridDim.x;
  int s0 = blockIdx.x * chunk;
  int s1 = min(S, s0 + chunk);
  float acc = 0.0f;
  for (int s = s0; s < s1; ++s) acc += X[(size_t)s * D_ + d];
  atomicAdd(&out128[d], acc * scale);
}

// out128[d] += scale * sum_n W[d, idx[n]]   (W row-major [128, V])
__global__ __launch_bounds__(128) void k_fm_gather_mean(
    const float* __restrict__ W, int V, const int* __restrict__ idx, int N,
    float scale, float* __restrict__ out128) {
  int d = threadIdx.x;
  int chunk = (N + gridDim.x - 1) / gridDim.x;
  int n0 = blockIdx.x * chunk;
  int n1 = min(N, n0 + chunk);
  float acc = 0.0f;
  for (int n = n0; n < n1; ++n) acc += W[(size_t)d * V + idx[n]];
  atomicAdd(&out128[d], acc * scale);
}

// h[n] = relu( sum_k x[k]*W[n,k] + b[n] ), single 128-wide block
__global__ __launch_bounds__(128) void k_dense_relu_1x128(
    const float* __restrict__ x, const float* __restrict__ W,
    const float* __restrict__ b, float* __restrict__ out) {
  int n = threadIdx.x;
  float acc = b[n];
  for (int k = 0; k < D_; ++k) acc += x[k] * W[n * D_ + k];
  out[n] = fmaxf(acc, 0.0f);
}

// w[s] = tanh( sum_k h[k] * hs[s,k] ), one wave per s (wave32 shuffle reduce)
__global__ __launch_bounds__(256) void k_attn_weights(
    const float* __restrict__ hs, const float* __restrict__ h,
    float* __restrict__ wout, int S) {
  int wave = blockIdx.x * 8 + (threadIdx.x >> 5);
  int lane = threadIdx.x & 31;
  if (wave >= S) return;
  const float* row = hs + (size_t)wave * D_;
  float acc = 0.0f;
#pragma unroll
  for (int i = 0; i < 4; ++i) {
    int k = lane + 32 * i;
    acc += row[k] * h[k];
  }
#pragma unroll
  for (int off = 16; off > 0; off >>= 1) acc += __shfl_xor(acc, off, 32);
  if (lane == 0) wout[wave] = tanhf(acc);
}

// out128[d] += scale * sum_s w[s] * hs[s,d]
__global__ __launch_bounds__(128) void k_weighted_colmean(
    const float* __restrict__ hs, const float* __restrict__ w, int S,
    float scale, float* __restrict__ out128) {
  int d = threadIdx.x;
  int chunk = (S + gridDim.x - 1) / gridDim.x;
  int s0 = blockIdx.x * chunk;
  int s1 = min(S, s0 + chunk);
  float acc = 0.0f;
  for (int s = s0; s < s1; ++s) acc += w[s] * hs[(size_t)s * D_ + d];
  atomicAdd(&out128[d], acc * scale);
}

// two 512x512 relu layers + 512->2 head, single block, one LDS buffer updated in place
__global__ __launch_bounds__(512) void k_final_mlp(
    const float* __restrict__ cat, const float* __restrict__ Wo2,
    const float* __restrict__ bo2, const float* __restrict__ Wi2,
    const float* __restrict__ bi2, float* __restrict__ out) {
  __shared__ float buf[512];
  int n = threadIdx.x;
  buf[n] = cat[n];
  __syncthreads();
  for (int j = 0; j < 2; ++j) {
    float acc = bo2[j * 512 + n];
    const float* wr = Wo2 + ((size_t)j * 512 + n) * 512;
    for (int k = 0; k < 512; ++k) acc += buf[k] * wr[k];
    __syncthreads();
    buf[n] = fmaxf(acc, 0.0f);
    __syncthreads();
  }
  if (n < 2) {
    float acc = bi2[n];
    const float* wr = Wi2 + n * 512;
    for (int k = 0; k < 512; ++k) acc += buf[k] * wr[k];
    out[n] = acc;
  }
}

// ---------------------------------------------------------------------------
extern "C" void kernel_launch(void* const* d_in, const int* in_sizes, int n_in,
                              void* d_out, int out_size, void* d_ws,
                              size_t ws_size, hipStream_t stream) {
  const int*   fingerprints = (const int*)d_in[0];
  const float* adjacency    = (const float*)d_in[1];
  const int*   words        = (const int*)d_in[2];
  const float* embed_fp     = (const float*)d_in[3];
  const float* embed_word   = (const float*)d_in[4];
  const float* Wg           = (const float*)d_in[5];
  const float* bg           = (const float*)d_in[6];
  const float* conv_w       = (const float*)d_in[7];
  const float* conv_b       = (const float*)d_in[8];
  const float* Wa           = (const float*)d_in[9];
  const float* ba           = (const float*)d_in[10];
  const float* Wo2          = (const float*)d_in[11];
  const float* bo2          = (const float*)d_in[12];
  const float* Wi2          = (const float*)d_in[13];
  const float* bi2          = (const float*)d_in[14];
  const float* Wc           = (const float*)d_in[15];
  const float* bc           = (const float*)d_in[16];
  // d_in[17] = v  (FM factors; interaction term is exactly zero)
  const float* Wp           = (const float*)d_in[18];
  const float* bp           = (const float*)d_in[19];
  // d_in[20] = wv (FM factors; interaction term is exactly zero)

  const int MAT = NA_ * D_;         // 512000
  float* f     = (float*)d_ws;
  float* xs    = f;                 // [4000,128]
  float* hs    = f + 512000;        // [4000,128]
  float* seqA  = f + 1024000;       // [4000,128]
  float* seqB  = f + 1536000;       // [4000,128]
  float* hsatt = f + 2048000;       // [4000,128]
  float* wts   = f + 2560000;       // [4000]
  float* cat   = f + 2564000;       // [512]
  float* hvec  = f + 2564512;       // [128]
  float* part  = f + 2564640;       // [SPLITK, 4000, 128] (split-K partials)

  const size_t need_splitk =
      (size_t)(2564640 + SPLITK_ * MAT) * sizeof(float);
  const bool use_splitk = (ws_size >= need_splitk);   // fixed per run

  const float invNA = 1.0f / (float)NA_;
  const float invNS = 1.0f / (float)NS_;

  // 1) embedding gathers
  k_gather_embed<<<(NA_ * D_ + 255) / 256, 256, 0, stream>>>(
      embed_fp, fingerprints, xs, NA_);
  k_gather_embed<<<(NS_ * D_ + 255) / 256, 256, 0, stream>>>(
      embed_word, words, seqA, NS_);

  // 2) GNN: 3x { hs = relu(xs@Wg^T+bg); xs += adj@hs }   (WMMA f32 path)
  dim3 ggrid((NA_ / 16 + 3) / 4, 2);      // 4 waves/block, N split by 2
  dim3 agrid((NA_ / 16 + 3) / 4, SPLITK_);
  for (int i = 0; i < 3; ++i) {
    k_gemm_xw_relu<<<ggrid, 128, 0, stream>>>(
        xs, Wg + (size_t)i * D_ * D_, bg + i * D_, hs, NA_);
    if (use_splitk) {
      k_gemm_adj_partial<<<agrid, 128, 0, stream>>>(
          adjacency, hs, part, NA_, NA_, NA_ / SPLITK_);
      k_reduce_splitk<<<(MAT + 255) / 256, 256, 0, stream>>>(xs, part, MAT);
    } else {
      k_gemm_adj<<<(NA_ / 16 + 3) / 4, 128, 0, stream>>>(
          adjacency, hs, xs, NA_, NA_);
    }
  }

  // 3) cat vector: [compound | fn_c | protein | fn_p]
  k_init_cat<<<1, 512, 0, stream>>>(cat, bc, bp);
  k_colmean<<<32, 128, 0, stream>>>(xs, NA_, invNA, cat + 0);
  k_fm_gather_mean<<<8, 128, 0, stream>>>(Wc, NF_, fingerprints, NA_, invNA,
                                          cat + 128);

  // 4) CNN stack (3x 11x11 same-padded conv + relu), ping-pong seqA/seqB
  dim3 cgrid(D_ / 32, NS_ / 8);
  k_conv11_relu<<<cgrid, 256, 0, stream>>>(seqA, seqB, conv_w, conv_b, 0, NS_);
  k_conv11_relu<<<cgrid, 256, 0, stream>>>(seqB, seqA, conv_w, conv_b, 1, NS_);
  k_conv11_relu<<<cgrid, 256, 0, stream>>>(seqA, seqB, conv_w, conv_b, 2, NS_);

  // 5) attention: hs_att = relu(seq@Wa^T+ba); h = relu(compound@Wa^T+ba)
  k_gemm_xw_relu<<<ggrid, 128, 0, stream>>>(seqB, Wa, ba, hsatt, NS_);
  k_dense_relu_1x128<<<1, 128, 0, stream>>>(cat, Wa, ba, hvec);
  k_attn_weights<<<(NS_ + 7) / 8, 256, 0, stream>>>(hsatt, hvec, wts, NS_);
  k_weighted_colmean<<<32, 128, 0, stream>>>(hsatt, wts, NS_, invNS, cat + 256);
  k_fm_gather_mean<<<8, 128, 0, stream>>>(Wp, NW_, words, NS_, invNS,
                                          cat + 384);

  // 6) output MLP head -> d_out[0:2]
  k_final_mlp<<<1, 512, 0, stream>>>(cat, Wo2, bo2, Wi2, bi2, (float*)d_out);
}